// CXTRNN_47717086659241
// MI455X (gfx1250) — compile-verified
//
#include <hip/hip_runtime.h>
#include <math.h>

// CXTRNN fully-fused persistent scan for MI455X (gfx1250, wave32).
//   T=2048 B=256 S=32 Z=16 R=16 H=128 Y=32
// 16 blocks (one per 16-row batch tile) x 96 threads (3 waves):
//   wave 0: recurrence  x_t = (1-a)x + a*( (tanh(x)@V * g) @ U^T + inp )   [critical path]
//   wave 1: inp_{t+1} = s@Win^T + b_in ; g_{t+1} = sigmoid(z@Wnm^T + b_nm) [prefetch, dbl-buffered LDS]
//   wave 2: y_{t-1}   = x_{t-1}@Wout^T + b_out -> HBM                      [drain, dbl-buffered LDS]
// All GEMMs use V_WMMA_F32_16X16X4_F32 (fp32 kept through the 2048-step chain).
// Wave 0 keeps state (64 VGPRs) and V / U^T weight fragments (128 VGPRs) in
// registers for the whole scan; tanh uses the hardware V_TANH_F32 trans op.

#define T_STEPS 2048
#define BATCH   256
#define DS_     32   // DIM_S
#define DZ_     16   // DIM_Z
#define RK_     16   // RANK
#define DH_     128  // DIM_HID
#define DY_     32   // DIM_Y
#define ALPHA   0.2f
#define BTILE   16
#define NTHREADS 96

typedef __attribute__((ext_vector_type(2))) float v2f;
typedef __attribute__((ext_vector_type(8))) float v8f;

__device__ __forceinline__ v8f wmma4(v2f a, v2f b, v8f c) {
    // D = A(16x4) * B(4x16) + C(16x16), fp32
    return __builtin_amdgcn_wmma_f32_16x16x4_f32(false, a, false, b, (short)0, c,
                                                 false, false);
}

__device__ __forceinline__ v8f v8zero() {
    v8f r = {0.f, 0.f, 0.f, 0.f, 0.f, 0.f, 0.f, 0.f};
    return r;
}

// Hardware transcendental tanh (V_TANH_F32, TRANS class -> co-issues with WMMA).
__device__ __forceinline__ float fast_tanh(float x) {
#if __has_builtin(__builtin_amdgcn_tanhf)
    return __builtin_amdgcn_tanhf(x);
#else
    float r;
    asm volatile("v_tanh_f32 %0, %1" : "=v"(r) : "v"(x));
    return r;
#endif
}

// A fragment: rows m=0..15 of a row-major [16 x ld] tile, K-cols k0..k0+3.
// Lane layout (ISA 7.12.2, 32-bit A 16x4): m = lane&15, k = 2*(lane>>4)+{0,1}.
__device__ __forceinline__ v2f frag_a(const float* p, int ld, int k0, int m, int kh) {
    const float* q = p + m * ld + k0 + 2 * kh;
    v2f r; r.x = q[0]; r.y = q[1]; return r;
}

// B fragment from a row-major [K x N] matrix (ld = N).
__device__ __forceinline__ v2f frag_b(const float* p, int ld, int k0, int n0, int nn, int kh) {
    v2f r;
    r.x = p[(k0 + 2 * kh)     * ld + n0 + nn];
    r.y = p[(k0 + 2 * kh + 1) * ld + n0 + nn];
    return r;
}

// B fragment = W^T where W is row-major [N x K] (ld = K): contiguous float2 per lane.
__device__ __forceinline__ v2f frag_bt(const float* p, int ld, int k0, int n0, int nn, int kh) {
    const float* q = p + (n0 + nn) * ld + k0 + 2 * kh;
    v2f r; r.x = q[0]; r.y = q[1]; return r;
}

__global__ __launch_bounds__(NTHREADS) void cxtrnn_fused_kernel(
    const float* __restrict__ s,    const float* __restrict__ z,
    const float* __restrict__ U,    const float* __restrict__ V,
    const float* __restrict__ Win,  const float* __restrict__ bin,
    const float* __restrict__ Wout, const float* __restrict__ bout,
    const float* __restrict__ Wnm,  const float* __restrict__ bnm,
    float* __restrict__ y)
{
    const int tid  = threadIdx.x;
    const int wave = tid >> 5;
    const int lane = tid & 31;
    const int m    = lane & 15;   // M / N index within tile
    const int nn   = m;
    const int kh   = lane >> 4;   // K-half select
    const int b0   = blockIdx.x * BTILE;

    // Critical-path weights + small state in LDS (~53 KB). Win/Wout stay in
    // global (L2-resident, touched only by non-critical waves 1/2).
    __shared__ float sU[DH_ * RK_];         // [H, R]
    __shared__ float sV[DH_ * RK_];         // [H, R]
    __shared__ float sWnm[RK_ * DZ_];       // [R, Z]
    __shared__ float s_bin[DH_];
    __shared__ float s_bout[DY_];
    __shared__ float s_bnm[RK_];
    __shared__ float inp_lds[2][BTILE * DH_];  // double-buffered inp_t  [16 x 128]
    __shared__ float g_lds[2][BTILE * RK_];    // double-buffered g_t    [16 x 16]
    __shared__ float x_lds[2][BTILE * DH_];    // double-buffered state  [16 x 128]
    __shared__ float p_lds[BTILE * RK_];       // wave0 layout-swap scratch

    for (int i = tid; i < DH_ * RK_; i += NTHREADS) { sU[i] = U[i]; sV[i] = V[i]; }
    for (int i = tid; i < RK_ * DZ_; i += NTHREADS) sWnm[i] = Wnm[i];
    for (int i = tid; i < DH_;       i += NTHREADS) s_bin[i]  = bin[i];
    for (int i = tid; i < DY_;       i += NTHREADS) s_bout[i] = bout[i];
    for (int i = tid; i < RK_;       i += NTHREADS) s_bnm[i]  = bnm[i];
    for (int i = tid; i < BTILE * DH_; i += NTHREADS) x_lds[1][i] = 0.f;  // x_{-1} = 0

    // wave 1: projections for step tt into buffer `buf`
    auto compute_inp_g = [&](int tt, int buf) {
        const float* s_t = s + ((size_t)tt * BATCH + b0) * DS_;
        const float* z_t = z + ((size_t)tt * BATCH + b0) * DZ_;
        // prefetch the s tile two steps ahead into the cache hierarchy
        if (tt + 2 < T_STEPS)
            __builtin_prefetch((const char*)(s + ((size_t)(tt + 2) * BATCH + b0) * DS_)
                               + lane * 64, 0, 3);
        // g = sigmoid(z @ Wnm^T + b_nm)
        v8f G = v8zero();
#pragma unroll
        for (int kk = 0; kk < DZ_ / 4; ++kk) {
            v2f a = frag_a(z_t, DZ_, 4 * kk, m, kh);
            v2f b = frag_bt(sWnm, DZ_, 4 * kk, 0, nn, kh);
            G = wmma4(a, b, G);
        }
        const float bn = s_bnm[nn];
#pragma unroll
        for (int v = 0; v < 8; ++v)
            g_lds[buf][(v + 8 * kh) * RK_ + nn] = 1.0f / (1.0f + __expf(-(G[v] + bn)));
        // inp = s @ Win^T + b_in
        v2f sa[DS_ / 4];
#pragma unroll
        for (int kk = 0; kk < DS_ / 4; ++kk) sa[kk] = frag_a(s_t, DS_, 4 * kk, m, kh);
#pragma unroll
        for (int j = 0; j < DH_ / 16; ++j) {
            v8f acc = v8zero();
#pragma unroll
            for (int kk = 0; kk < DS_ / 4; ++kk) {
                v2f b = frag_bt(Win, DS_, 4 * kk, 16 * j, nn, kh);
                acc = wmma4(sa[kk], b, acc);
            }
            const float bi = s_bin[16 * j + nn];
#pragma unroll
            for (int v = 0; v < 8; ++v)
                inp_lds[buf][(v + 8 * kh) * DH_ + 16 * j + nn] = acc[v] + bi;
        }
    };

    // wave 2: y_tt = x_lds[buf] @ Wout^T + b_out
    auto compute_y = [&](int tt, int buf) {
        float* y_t = y + ((size_t)tt * BATCH + b0) * DY_;
#pragma unroll
        for (int j = 0; j < DY_ / 16; ++j) {
            v8f acc = v8zero();
#pragma unroll
            for (int kk = 0; kk < DH_ / 4; ++kk) {
                v2f a = frag_a(x_lds[buf], DH_, 4 * kk, m, kh);
                v2f b = frag_bt(Wout, DH_, 4 * kk, 16 * j, nn, kh);
                acc = wmma4(a, b, acc);
            }
            const float bo = s_bout[16 * j + nn];
#pragma unroll
            for (int v = 0; v < 8; ++v)
                y_t[(v + 8 * kh) * DY_ + 16 * j + nn] = acc[v] + bo;
        }
    };

    __syncthreads();

    // Loop-invariant weight fragments for the critical wave, held in VGPRs for
    // the entire 2048-step scan (the per-step barrier/memory clobber would
    // otherwise force LDS reloads every iteration).
    v2f Vb[DH_ / 4];    // B-frags of V   (B = V,    [128 x 16])
    v2f Ub[32];         // B-frags of U^T (B = U^T,  [16 x 128]) : [j][kk]
    v8f X[DH_ / 16];    // recurrent state, C/D fragments
#pragma unroll
    for (int j = 0; j < DH_ / 16; ++j) X[j] = v8zero();

    if (wave == 0) {
        asm volatile("s_setprio 3");           // win SIMD arbitration
#pragma unroll
        for (int kk = 0; kk < DH_ / 4; ++kk)
            Vb[kk] = frag_b(sV, RK_, 4 * kk, 0, nn, kh);
#pragma unroll
        for (int j = 0; j < DH_ / 16; ++j)
#pragma unroll
            for (int kk = 0; kk < RK_ / 4; ++kk)
                Ub[j * 4 + kk] = frag_bt(sU, RK_, 4 * kk, 16 * j, nn, kh);
    } else if (wave == 1) {
        compute_inp_g(0, 0);                   // prime the pipeline
    }
    __syncthreads();

    for (int t = 0; t < T_STEPS; ++t) {
        const int cur = t & 1, nxt = cur ^ 1;
        // entry invariant: inp_lds/g_lds[cur] = step-t values; x_lds[nxt] = x_{t-1}
        if (wave == 0) {
            // P = tanh(x_{t-1}) @ V        (16x128 * 128x16)
            v8f P = v8zero();
#pragma unroll
            for (int kk = 0; kk < DH_ / 4; ++kk) {
                v2f a = frag_a(x_lds[nxt], DH_, 4 * kk, m, kh);
                a.x = fast_tanh(a.x); a.y = fast_tanh(a.y);
                P = wmma4(a, Vb[kk], P);
            }
            // P *= g (identical C/D layouts)
#pragma unroll
            for (int v = 0; v < 8; ++v)
                P[v] *= g_lds[cur][(v + 8 * kh) * RK_ + nn];
            // layout swap C/D -> A via LDS (same-wave LDS ops are in-order)
#pragma unroll
            for (int v = 0; v < 8; ++v)
                p_lds[(v + 8 * kh) * RK_ + nn] = P[v];
            asm volatile("s_wait_dscnt 0" ::: "memory");
            // X = (1-a)X + a*( P @ U^T + inp )   (inp folded into WMMA C)
#pragma unroll
            for (int j = 0; j < DH_ / 16; ++j) {
                v8f acc;
#pragma unroll
                for (int v = 0; v < 8; ++v)
                    acc[v] = inp_lds[cur][(v + 8 * kh) * DH_ + 16 * j + nn];
#pragma unroll
                for (int kk = 0; kk < RK_ / 4; ++kk) {
                    v2f a = frag_a(p_lds, RK_, 4 * kk, m, kh);
                    acc = wmma4(a, Ub[j * 4 + kk], acc);
                }
#pragma unroll
                for (int v = 0; v < 8; ++v) {
                    X[j][v] = (1.0f - ALPHA) * X[j][v] + ALPHA * acc[v];
                    x_lds[cur][(v + 8 * kh) * DH_ + 16 * j + nn] = X[j][v];
                }
            }
        } else if (wave == 1) {
            if (t + 1 < T_STEPS) compute_inp_g(t + 1, nxt);
        } else {
            if (t >= 1) compute_y(t - 1, nxt);   // x_lds[nxt] holds x_{t-1}
        }
        __syncthreads();
    }
    if (wave == 2) compute_y(T_STEPS - 1, (T_STEPS - 1) & 1);  // drain last y
}

extern "C" void kernel_launch(void* const* d_in, const int* in_sizes, int n_in,
                              void* d_out, int out_size, void* d_ws, size_t ws_size,
                              hipStream_t stream) {
    (void)in_sizes; (void)n_in; (void)out_size; (void)d_ws; (void)ws_size;
    const float* s    = (const float*)d_in[0];
    const float* z    = (const float*)d_in[1];
    const float* U    = (const float*)d_in[2];
    const float* V    = (const float*)d_in[3];
    const float* Win  = (const float*)d_in[4];
    const float* bin  = (const float*)d_in[5];
    const float* Wout = (const float*)d_in[6];
    const float* bout = (const float*)d_in[7];
    const float* Wnm  = (const float*)d_in[8];
    const float* bnm  = (const float*)d_in[9];
    float* yout = (float*)d_out;

    dim3 grid(BATCH / BTILE);   // 16 blocks, one per 16-row batch tile
    dim3 block(NTHREADS);       // 3 wave32s: recurrence / prefetch / drain
    hipLaunchKernelGGL(cxtrnn_fused_kernel, grid, block, 0, stream,
                       s, z, U, V, Win, bin, Wout, bout, Wnm, bnm, yout);
}